// CTCLayer_80393197847138
// MI455X (gfx1250) — compile-verified
//
#include <hip/hip_runtime.h>

typedef __attribute__((ext_vector_type(2))) float v2f;
typedef __attribute__((ext_vector_type(8))) float v8f;

#define B_   256
#define T_   1024
#define C_   96
#define L_   128
#define S_   257      // 2L+1 extended states
#define SP   272      // padded S: 17 tiles * 16
#define BLANK 95
#define NEGV (-1e30f)
#define EPS_  1e-7f
#define TCH  16       // timesteps per chunk
#define NTHREADS 288  // 9 wave32

// logaddexp: max + log1p(exp(min-max)); NEG-safe (exp(0) path adds ~0.69 to -1e30 -> -1e30)
__device__ __forceinline__ float lae(float x, float y) {
    float m = fmaxf(x, y);
    float d = fminf(x, y) - m;          // <= 0
    return m + __logf(1.0f + __expf(d));
}

__global__ __launch_bounds__(NTHREADS)
void ctc_fwd_kernel(const int* __restrict__ y_true,
                    const float* __restrict__ y_pred,
                    float* __restrict__ out)
{
    __shared__ int   labS[L_];
    __shared__ float logpS[TCH * C_];   // [t][c] log(p+eps) for current 16-step chunk
    __shared__ float lpS[TCH * SP];     // [t][s] gathered emission log-probs
    __shared__ float alphaS[2][SP];     // double-buffered forward variables

    const int blk  = blockIdx.x;        // sample index
    const int tid  = threadIdx.x;
    const int lane = tid & 31;
    const int wave = tid >> 5;          // 0..8

    // ---- labels to LDS ----
    if (tid < L_) labS[tid] = y_true[blk * L_ + tid];
    __syncthreads();

    // ---- per-state skip flag (thread tid owns state s = tid) ----
    const int s = tid;
    bool skip = false;
    if (s >= 2 && s < S_) {
        int es  = (s & 1)       ? labS[s >> 1]       : BLANK;
        int es2 = ((s - 2) & 1) ? labS[(s - 2) >> 1] : BLANK;
        skip = (es != BLANK) && (es != es2);
    }

    // ---- per-lane one-hot metadata for WMMA B fragments ----
    // wave w computes s-tiles w and w+9; B lane ln: n = ln&15, kh = ln>>4, K = v + 2*kh
    int colExt0, colExt1;
    {
        int n  = lane & 15;
        int c0 = wave * 16 + n;          // state column, tile wave
        int c1 = (wave + 9) * 16 + n;    // state column, tile wave+9 (>=SP for wave 8: unused)
        int i0 = c0 >> 1, i1 = c1 >> 1;
        colExt0 = (c0 & 1) ? ((i0 < L_) ? labS[i0] : BLANK) : BLANK;
        colExt1 = (c1 & 1) ? ((i1 < L_) ? labS[i1] : BLANK) : BLANK;
    }

    const float* ypB = y_pred + (size_t)blk * T_ * C_;

    int p = 0;
    for (int t0 = 0; t0 < T_; t0 += TCH) {
        // ---- prefetch next chunk's 16x96 block (96 x 64B lines) ----
        if (t0 + TCH < T_ && tid < 96) {
            __builtin_prefetch(ypB + (size_t)(t0 + TCH) * C_ + tid * 16, 0, 1);
        }

        // ---- load chunk (contiguous 1536 floats) + log into LDS ----
        {
            const float4* src = (const float4*)(ypB + (size_t)t0 * C_);
            for (int i = tid; i < (TCH * C_) / 4; i += NTHREADS) {
                float4 v = src[i];
                float* dst = &logpS[i * 4];
                dst[0] = __logf(v.x + EPS_);
                dst[1] = __logf(v.y + EPS_);
                dst[2] = __logf(v.z + EPS_);
                dst[3] = __logf(v.w + EPS_);
            }
        }
        __syncthreads();

        // ---- exact one-hot gather as f32 WMMA:
        //      lp(16t x 16s) = sum_k logp(16t x 4k) * onehot(4k x 16s) ----
        {
            const int m  = lane & 15;     // timestep row for A
            const int kh = lane >> 4;     // lane-half -> K = v + 2*kh
            v8f acc0 = {};
            v8f acc1 = {};
            for (int kc = 0; kc < C_; kc += 4) {
                v2f av, bv0, bv1;
                int k0 = kc + 2 * kh;
                av[0]  = logpS[m * C_ + k0 + 0];
                av[1]  = logpS[m * C_ + k0 + 1];
                bv0[0] = (colExt0 == k0 + 0) ? 1.0f : 0.0f;
                bv0[1] = (colExt0 == k0 + 1) ? 1.0f : 0.0f;
                bv1[0] = (colExt1 == k0 + 0) ? 1.0f : 0.0f;
                bv1[1] = (colExt1 == k0 + 1) ? 1.0f : 0.0f;
                acc0 = __builtin_amdgcn_wmma_f32_16x16x4_f32(
                    false, av, false, bv0, (short)0, acc0, false, false);
                acc1 = __builtin_amdgcn_wmma_f32_16x16x4_f32(
                    false, av, false, bv1, (short)0, acc1, false, false);
            }
            // D layout: VGPR r -> t = r + 8*kh, s = tile*16 + (lane&15)
            int n   = lane & 15;
            int sc0 = wave * 16 + n;
            int sc1 = (wave + 9) * 16 + n;
            #pragma unroll
            for (int r = 0; r < 8; ++r) {
                int tt = r + 8 * kh;
                lpS[tt * SP + sc0] = acc0[r];
                if (sc1 < SP) lpS[tt * SP + sc1] = acc1[r];
            }
        }
        __syncthreads();

        // ---- 16 serial scan steps ----
        for (int tl = 0; tl < TCH; ++tl) {
            const int t = t0 + tl;
            float* prev = alphaS[p];
            float* cur  = alphaS[p ^ 1];
            if (s < S_) {
                float lp = lpS[tl * SP + s];
                float v;
                if (t == 0) {
                    v = (s < 2) ? lp : NEGV;
                } else {
                    float a0 = prev[s];
                    float a1 = (s >= 1) ? prev[s - 1] : NEGV;
                    v = lae(a0, a1);
                    if (skip) v = lae(v, prev[s - 2]);
                    v += lp;
                }
                cur[s] = v;
            }
            __syncthreads();
            p ^= 1;
        }
    }

    if (tid == 0) {
        float* fin = alphaS[p];
        out[blk] = -lae(fin[S_ - 1], fin[S_ - 2]);
    }
}

extern "C" void kernel_launch(void* const* d_in, const int* in_sizes, int n_in,
                              void* d_out, int out_size, void* d_ws, size_t ws_size,
                              hipStream_t stream) {
    (void)in_sizes; (void)n_in; (void)out_size; (void)d_ws; (void)ws_size;
    const int*   y_true = (const int*)d_in[0];
    const float* y_pred = (const float*)d_in[1];
    float*       out    = (float*)d_out;
    ctc_fwd_kernel<<<B_, NTHREADS, 0, stream>>>(y_true, y_pred, out);
}